// ROIPooling_5746666242435
// MI455X (gfx1250) — compile-verified
//
#include <hip/hip_runtime.h>
#include <hip/hip_bf16.h>
#include <stdint.h>

#define P 7
#define C_DIM 256
#define H_DIM 64
#define W_DIM 64
#define CH 4            // channels per block
#define NTHREADS 256

typedef unsigned int v4u __attribute__((ext_vector_type(4)));
typedef int          v8i __attribute__((ext_vector_type(8)));
typedef int          v4i __attribute__((ext_vector_type(4)));

__global__ __launch_bounds__(NTHREADS) void roi_pool_tdm_kernel(
    const float* __restrict__ features, const float* __restrict__ rois,
    float* __restrict__ out, float* __restrict__ out_bid) {
  __shared__ float smem[CH * H_DIM * W_DIM];   // 64 KB worst-case tile

  const int groups_per_roi = C_DIM / CH;
  const int n  = blockIdx.x / groups_per_roi;
  const int g  = blockIdx.x % groups_per_roi;
  const int c0 = g * CH;

  const float* r = rois + (size_t)n * 5;
  const int b  = (int)r[0];
  const int px = (int)rintf(r[1] * 0.0625f);
  const int py = (int)rintf(r[2] * 0.0625f);
  const int qx = (int)rintf(r[3] * 0.0625f);
  const int qy = (int)rintf(r[4] * 0.0625f);

  int width  = qx - px + 1; if (width  < 1) width  = 1;
  int height = qy - py + 1; if (height < 1) height = 1;

#if __has_builtin(__builtin_amdgcn_tensor_load_to_lds)
  // Uniform (SGPR) per-wave guard: only wave 0's first-lane tid is 0, so
  // waves 1..7 take a scalar branch around the TDM issue (TDM ignores EXEC,
  // so an exec-mask guard alone would not prevent redundant per-wave DMAs).
  if (__builtin_amdgcn_readfirstlane(threadIdx.x) == 0) {
    const uint64_t gaddr = (uint64_t)(uintptr_t)(features
        + ((size_t)b * C_DIM + c0) * (size_t)(H_DIM * W_DIM)
        + (size_t)py * W_DIM + (size_t)px);
    const uint32_t lds = (uint32_t)(uintptr_t)(void*)smem;  // low 32b = LDS offset

    v4u g0; v8i g1; v4i g2; v4i g3;
    // ---- D# group 0: count=1 (valid), lds_addr, global_addr[56:0], type=2
    g0[0] = 1u;
    g0[1] = lds;
    g0[2] = (uint32_t)gaddr;
    g0[3] = (uint32_t)((gaddr >> 32) & 0x01FFFFFFu) | (2u << 30);
    // ---- D# group 1
    g1[0] = (int)(2u << 16);                                   // data_size = 4B, no pad/iter/mask
    g1[1] = (int)(((uint32_t)width & 0xFFFFu) << 16);          // tensor_dim0[15:0] @ bits 63:48
    g1[2] = (int)((((uint32_t)width >> 16) & 0xFFFFu)          // tensor_dim0[31:16]
                  | (((uint32_t)height & 0xFFFFu) << 16));     // tensor_dim1[15:0]
    g1[3] = (int)(((uint32_t)height >> 16)                     // tensor_dim1[31:16]
                  | ((uint32_t)width << 16));                  // tile_dim0
    g1[4] = (int)((uint32_t)height | ((uint32_t)CH << 16));    // tile_dim1 | tile_dim2
    g1[5] = W_DIM;                                             // tensor_dim0_stride (elems)
    g1[6] = (int)(((uint32_t)(H_DIM * W_DIM) & 0xFFFFu) << 16);// tensor_dim1_stride[15:0]
    g1[7] = (int)((uint32_t)(H_DIM * W_DIM) >> 16);            // tensor_dim1_stride[47:16]
    // ---- D# group 2: tensor_dim2 = CH, tensor_dim2_stride = H*W, tile_dim3 = 0
    g2[0] = CH; g2[1] = 0; g2[2] = H_DIM * W_DIM; g2[3] = 0;
    // ---- D# group 3: unused dims
    g3[0] = 0; g3[1] = 0; g3[2] = 0; g3[3] = 0;

#if defined(__clang_major__) && (__clang_major__ >= 23)
    // clang-23 / therock-10.0 lane: 6-arg form, extra int32x8 = VADDR4 (unused)
    v8i g4 = {0, 0, 0, 0, 0, 0, 0, 0};
    __builtin_amdgcn_tensor_load_to_lds(g0, g1, g2, g3, g4, 0);
#else
    // ROCm 7.2 / clang-22 lane: 5-arg form
    __builtin_amdgcn_tensor_load_to_lds(g0, g1, g2, g3, 0);
#endif
#if __has_builtin(__builtin_amdgcn_s_wait_tensorcnt)
    __builtin_amdgcn_s_wait_tensorcnt(0);
#endif
  }
  __syncthreads();
#else
  // Fallback: cooperative staged loads (same LDS layout)
  for (int idx = threadIdx.x; idx < CH * height * width; idx += NTHREADS) {
    int cc  = idx / (height * width);
    int rem = idx % (height * width);
    int yy = rem / width, xx = rem % width;
    smem[idx] = features[((size_t)b * C_DIM + c0 + cc) * (size_t)(H_DIM * W_DIM)
                         + (size_t)(py + yy) * W_DIM + (size_t)(px + xx)];
  }
  __syncthreads();
#endif

  // ---- per-bin closed-form rectangles, mirroring _axis_bins semantics ----
  const int psx   = (width  + P - 1) / P;
  const int pad0x = (psx * P - width)  >> 1;
  const int psy   = (height + P - 1) / P;
  const int pad0y = (psy * P - height) >> 1;

  const int t = threadIdx.x;
  if (t < CH * P * P) {
    const int cc  = t / (P * P);
    const int bin = t % (P * P);
    const int by = bin / P, bx = bin % P;

    int xs = bx * psx - pad0x, xe = xs + psx - 1;
    int ys = by * psy - pad0y, ye = ys + psy - 1;
    const bool pad = (xs < 0) | (xe > width - 1) | (ys < 0) | (ye > height - 1);
    if (xs < 0) xs = 0;
    if (xe > width - 1) xe = width - 1;
    if (ys < 0) ys = 0;
    if (ye > height - 1) ye = height - 1;

    float val;
    if (xs > xe || ys > ye) {
      val = 0.0f;                                  // fully-padded bin -> 0
    } else {
      val = -__builtin_inff();
      const float* base = smem + (size_t)cc * height * width;
      for (int yy = ys; yy <= ye; ++yy) {
        const float* row = base + (size_t)yy * width;
        for (int xx = xs; xx <= xe; ++xx)
          val = fmaxf(val, row[xx]);
      }
      if (pad) val = fmaxf(val, 0.0f);             // bin overlaps zero padding
    }
    out[(((size_t)n * C_DIM + c0 + cc) * P + by) * P + bx] = val;
  }

  // second tuple output: batch index per ROI (promoted to f32 in concat)
  if (threadIdx.x == NTHREADS - 1 && g == 0) out_bid[n] = (float)b;
}

extern "C" void kernel_launch(void* const* d_in, const int* in_sizes, int n_in,
                              void* d_out, int out_size, void* d_ws, size_t ws_size,
                              hipStream_t stream) {
  const float* features = (const float*)d_in[0];
  const float* rois     = (const float*)d_in[1];
  float* out = (float*)d_out;
  const int n_rois = in_sizes[1] / 5;
  float* out_bid = out + (size_t)n_rois * C_DIM * P * P;

  dim3 grid(n_rois * (C_DIM / CH));
  roi_pool_tdm_kernel<<<grid, NTHREADS, 0, stream>>>(features, rois, out, out_bid);
}